// QuantumSuperposition_17849884082754
// MI455X (gfx1250) — compile-verified
//
#include <hip/hip_runtime.h>
#include <hip/hip_bf16.h>
#include <math.h>

// ---------------- problem constants ----------------
#define NROWS 1024           // B*T = 16*64
#define DMOD  128            // D
#define DINN  256            // D_INNER
#define DSTATE 16
#define DTRANK 8
#define C1DIM 64
#define NCDIM 4
#define DBC_LD 48            // padded leading dim for the 40-wide dbc matrix

typedef float v2f __attribute__((ext_vector_type(2)));
typedef float v8f __attribute__((ext_vector_type(8)));

// ---------------------------------------------------------------------------
// Generic f32 WMMA GEMM:  C[M,N] = act(A[M,K] @ B[K,N] + bias)
// One wave (32 lanes) computes a 16x32 tile (two 16x16 WMMA col-tiles sharing
// one A fragment) using V_WMMA_F32_16X16X4_F32.  block=(32,4): 4 row-tiles.
// K must be a multiple of 4.
// A-frag (16x4, ISA 7.12.2):  lane L, comp j -> A[L%16][2*(L/16)+j]
// B-frag (4x16):              lane L, comp j -> B[2*(L/16)+j][L%16]
// C/D  (16x16):               vgpr r, lane L -> C[r+8*(L/16)][L%16]
// Column guard is branchless (clamped index + 0/1 mask) so EXEC stays all-1s
// across every WMMA issue.
// ---------------------------------------------------------------------------
__global__ void gemm16_wmma_f32(const float* __restrict__ A, int lda,
                                const float* __restrict__ Bm, int ldb,
                                const float* __restrict__ bias,
                                float* __restrict__ C, int ldc,
                                int N, int K, int relu_act)
{
    const int lane  = threadIdx.x;          // 0..31 (one wave)
    const int half  = lane >> 4;            // 0 or 1
    const int l16   = lane & 15;
    const int row0  = (blockIdx.y * 4 + threadIdx.y) * 16;
    const int col0  = blockIdx.x * 32;

    const int cc0   = col0 + l16;
    const int cc1   = cc0 + 16;
    const float m0  = (cc0 < N) ? 1.0f : 0.0f;    // branchless col masks
    const float m1  = (cc1 < N) ? 1.0f : 0.0f;
    const int cs0   = (cc0 < N) ? cc0 : (N - 1);  // clamped (always legal) cols
    const int cs1   = (cc1 < N) ? cc1 : (N - 1);

    v8f acc0 = {};                          // two f32 accumulators
    v8f acc1 = {};
    const float* arow = A + (size_t)(row0 + l16) * lda;

    for (int k = 0; k < K; k += 4) {
        const int ka = k + 2 * half;
        v2f a;
        a.x = arow[ka];
        a.y = arow[ka + 1];
        const float* br0 = Bm + (size_t)ka * ldb;
        const float* br1 = Bm + (size_t)(ka + 1) * ldb;
        v2f b0, b1;
        b0.x = br0[cs0] * m0;
        b0.y = br1[cs0] * m0;
        b1.x = br0[cs1] * m1;
        b1.y = br1[cs1] * m1;
        acc0 = __builtin_amdgcn_wmma_f32_16x16x4_f32(
                   false, a, false, b0, (short)0, acc0, false, false);
        acc1 = __builtin_amdgcn_wmma_f32_16x16x4_f32(
                   false, a, false, b1, (short)0, acc1, false, false);
    }

    // epilogue: bias + optional relu, guarded stores (divergence after WMMAs)
    if (cc0 < N) {
        const float bv = bias ? bias[cc0] : 0.0f;
        #pragma unroll
        for (int r = 0; r < 8; ++r) {
            float v = acc0[r] + bv;
            if (relu_act) v = fmaxf(v, 0.0f);
            C[(size_t)(row0 + r + 8 * half) * ldc + cc0] = v;
        }
    }
    if (cc1 < N) {
        const float bv = bias ? bias[cc1] : 0.0f;
        #pragma unroll
        for (int r = 0; r < 8; ++r) {
            float v = acc1[r] + bv;
            if (relu_act) v = fmaxf(v, 0.0f);
            C[(size_t)(row0 + r + 8 * half) * ldc + cc1] = v;
        }
    }
}

// ---------------------------------------------------------------------------
// vr[b,t,d] = sum_m softmax_m(||reps_m||) * (reps_m[d] / max(||reps_m||,1e-12))^2
// one block per row; 128 threads (one per d)
// ---------------------------------------------------------------------------
__global__ void fuse_vr_kernel(const float* __restrict__ r0,
                               const float* __restrict__ r1,
                               const float* __restrict__ r2,
                               float* __restrict__ vr)
{
    const int row = blockIdx.x;
    const int d   = threadIdx.x;
    __shared__ float sm0[128], sm1[128], sm2[128];

    const float v0 = r0[row * DMOD + d];
    const float v1 = r1[row * DMOD + d];
    const float v2 = r2[row * DMOD + d];
    sm0[d] = v0 * v0; sm1[d] = v1 * v1; sm2[d] = v2 * v2;
    __syncthreads();
    for (int s = 64; s > 0; s >>= 1) {
        if (d < s) { sm0[d] += sm0[d + s]; sm1[d] += sm1[d + s]; sm2[d] += sm2[d + s]; }
        __syncthreads();
    }
    const float n0 = sqrtf(sm0[0]), n1 = sqrtf(sm1[0]), n2 = sqrtf(sm2[0]);
    const float mx = fmaxf(n0, fmaxf(n1, n2));
    const float e0 = expf(n0 - mx), e1 = expf(n1 - mx), e2 = expf(n2 - mx);
    const float inv = 1.0f / (e0 + e1 + e2);
    const float c0 = fmaxf(n0, 1e-12f), c1 = fmaxf(n1, 1e-12f), c2 = fmaxf(n2, 1e-12f);
    const float a0 = v0 / c0, a1 = v1 / c1, a2 = v2 / c2;
    vr[row * DMOD + d] = e0 * inv * a0 * a0 + e1 * inv * a1 * a1 + e2 * inv * a2 * a2;
}

// ---------------------------------------------------------------------------
// split xz -> x = silu(x*conv_w + conv_b), gate = silu(zg)
// ---------------------------------------------------------------------------
__global__ void xz_post_kernel(const float* __restrict__ xz,
                               const float* __restrict__ conv_w,
                               const float* __restrict__ conv_b,
                               float* __restrict__ xbuf,
                               float* __restrict__ gate)
{
    const int row = blockIdx.x;
    const int j   = threadIdx.x;     // 0..255
    const float xv = xz[(size_t)row * (2 * DINN) + j];
    const float zg = xz[(size_t)row * (2 * DINN) + DINN + j];
    const float t  = xv * conv_w[j] + conv_b[j];
    xbuf[(size_t)row * DINN + j] = t / (1.0f + expf(-t));
    gate[(size_t)row * DINN + j] = zg / (1.0f + expf(-zg));
}

// ---------------------------------------------------------------------------
// bc[row] = sum_s Bc[row,s]*Cc[row,s]   (cols 8..23 and 24..39 of padded dbc)
// ---------------------------------------------------------------------------
__global__ void bc_kernel(const float* __restrict__ dbc, float* __restrict__ bcv)
{
    const int row = blockIdx.x * blockDim.x + threadIdx.x;
    if (row < NROWS) {
        float s = 0.0f;
        #pragma unroll
        for (int i = 0; i < DSTATE; ++i)
            s += dbc[(size_t)row * DBC_LD + DTRANK + i]
               * dbc[(size_t)row * DBC_LD + DTRANK + DSTATE + i];
        bcv[row] = s;
    }
}

// ---------------------------------------------------------------------------
// g = (softplus(dpre)*x*bc + D_ssm*x) * gate   (dpre already has +b_dt)
// ---------------------------------------------------------------------------
__global__ void ssm_gate_kernel(const float* __restrict__ dpre,
                                const float* __restrict__ xbuf,
                                const float* __restrict__ bcv,
                                const float* __restrict__ D_ssm,
                                const float* __restrict__ gate,
                                float* __restrict__ gbuf)
{
    const int row = blockIdx.x;
    const int j   = threadIdx.x;     // 0..255
    const float dp = dpre[(size_t)row * DINN + j];
    const float delta = (dp > 20.0f) ? dp : log1pf(expf(dp));
    const float x = xbuf[(size_t)row * DINN + j];
    const float y = delta * x * bcv[row] + D_ssm[j] * x;
    gbuf[(size_t)row * DINN + j] = y * gate[(size_t)row * DINN + j];
}

// ---------------------------------------------------------------------------
// normalize kernel rows and transpose: krT[d,k], kiT[d,k]
// one block per k; 256 threads do the Frobenius norm over 256 elems
// ---------------------------------------------------------------------------
__global__ void kernel_norm_kernel(const float* __restrict__ kern,
                                   float* __restrict__ krT,
                                   float* __restrict__ kiT)
{
    const int k = blockIdx.x;
    const int t = threadIdx.x;       // 0..255
    __shared__ float sm[256];
    const float v = kern[(size_t)k * 256 + t];
    sm[t] = v * v;
    __syncthreads();
    for (int s = 128; s > 0; s >>= 1) {
        if (t < s) sm[t] += sm[t + s];
        __syncthreads();
    }
    const float n = fmaxf(sqrtf(sm[0]), 1e-12f);
    if (t < DMOD) {
        krT[(size_t)t * DMOD + k] = kern[(size_t)k * 256 + t * 2 + 0] / n;
        kiT[(size_t)t * DMOD + k] = kern[(size_t)k * 256 + t * 2 + 1] / n;
    }
}

// probs = pr^2 + pi^2
__global__ void probs_sq_kernel(const float* __restrict__ pr,
                                const float* __restrict__ pi,
                                float* __restrict__ probs)
{
    const int i = blockIdx.x * blockDim.x + threadIdx.x;
    if (i < NROWS * DMOD) probs[i] = pr[i] * pr[i] + pi[i] * pi[i];
}

// out = log_softmax(tanh(f1 @ W_fc2 + b_fc2))   (one thread per row)
__global__ void head_kernel(const float* __restrict__ f1,
                            const float* __restrict__ W_fc2,
                            const float* __restrict__ b_fc2,
                            float* __restrict__ out)
{
    const int row = blockIdx.x * blockDim.x + threadIdx.x;
    if (row >= NROWS) return;
    float o[NCDIM];
    #pragma unroll
    for (int c = 0; c < NCDIM; ++c) {
        float s = b_fc2[c];
        for (int j = 0; j < C1DIM; ++j)
            s += f1[(size_t)row * C1DIM + j] * W_fc2[j * NCDIM + c];
        o[c] = tanhf(s);
    }
    float m = o[0];
    #pragma unroll
    for (int c = 1; c < NCDIM; ++c) m = fmaxf(m, o[c]);
    float se = 0.0f;
    #pragma unroll
    for (int c = 0; c < NCDIM; ++c) se += expf(o[c] - m);
    const float lse = m + logf(se);
    #pragma unroll
    for (int c = 0; c < NCDIM; ++c) out[(size_t)row * NCDIM + c] = o[c] - lse;
}

// ---------------- workspace layout (float offsets) ----------------
#define OFF_REPS0  ((size_t)0)
#define OFF_REPS1  (OFF_REPS0 + (size_t)NROWS * DMOD)
#define OFF_REPS2  (OFF_REPS1 + (size_t)NROWS * DMOD)
#define OFF_VR     (OFF_REPS2 + (size_t)NROWS * DMOD)
#define OFF_XZ     (OFF_VR    + (size_t)NROWS * DMOD)          // 1024x512
#define OFF_XBUF   (OFF_XZ    + (size_t)NROWS * 2 * DINN)
#define OFF_GATE   (OFF_XBUF  + (size_t)NROWS * DINN)
#define OFF_DBC    (OFF_GATE  + (size_t)NROWS * DINN)          // 1024x48 (padded)
#define OFF_BCV    (OFF_DBC   + (size_t)NROWS * DBC_LD)
#define OFF_DPRE   (OFF_BCV   + (size_t)NROWS)
#define OFF_GBUF   (OFF_DPRE  + (size_t)NROWS * DINN)
#define OFF_H      (OFF_GBUF  + (size_t)NROWS * DINN)
#define OFF_KRT    (OFF_H     + (size_t)NROWS * DMOD)
#define OFF_KIT    (OFF_KRT   + (size_t)DMOD * DMOD)
#define OFF_PR     (OFF_KIT   + (size_t)DMOD * DMOD)
#define OFF_PI     (OFF_PR    + (size_t)NROWS * DMOD)
#define OFF_PROBS  (OFF_PI    + (size_t)NROWS * DMOD)
#define OFF_F1     (OFF_PROBS + (size_t)NROWS * DMOD)

static inline void launch_gemm(const float* A, int lda, const float* B, int ldb,
                               const float* bias, float* C, int ldc,
                               int M, int N, int K, int relu_act, hipStream_t stream)
{
    dim3 block(32, 4);
    dim3 grid((N + 31) / 32, M / 64);
    gemm16_wmma_f32<<<grid, block, 0, stream>>>(A, lda, B, ldb, bias, C, ldc, N, K, relu_act);
}

extern "C" void kernel_launch(void* const* d_in, const int* in_sizes, int n_in,
                              void* d_out, int out_size, void* d_ws, size_t ws_size,
                              hipStream_t stream)
{
    const float* text    = (const float*)d_in[0];
    const float* audio   = (const float*)d_in[1];
    const float* visual  = (const float*)d_in[2];
    /* d_in[3] beats: phase cancels on the diagonal -> unused */
    const float* W_text  = (const float*)d_in[4];
    const float* b_text  = (const float*)d_in[5];
    const float* W_audio = (const float*)d_in[6];
    const float* b_audio = (const float*)d_in[7];
    const float* W_vis   = (const float*)d_in[8];
    const float* b_vis   = (const float*)d_in[9];
    /* d_in[10..11] W_beats/b_beats unused */
    const float* W_in    = (const float*)d_in[12];
    const float* conv_w  = (const float*)d_in[13];
    const float* conv_b  = (const float*)d_in[14];
    const float* W_xproj = (const float*)d_in[15];
    const float* W_dt    = (const float*)d_in[16];
    const float* b_dt    = (const float*)d_in[17];
    const float* D_ssm   = (const float*)d_in[18];
    const float* W_out   = (const float*)d_in[19];
    const float* kern    = (const float*)d_in[20];
    const float* W_fc1   = (const float*)d_in[21];
    const float* b_fc1   = (const float*)d_in[22];
    const float* W_fc2   = (const float*)d_in[23];
    const float* b_fc2   = (const float*)d_in[24];

    float* ws = (float*)d_ws;
    float* reps0 = ws + OFF_REPS0;
    float* reps1 = ws + OFF_REPS1;
    float* reps2 = ws + OFF_REPS2;
    float* vr    = ws + OFF_VR;
    float* xz    = ws + OFF_XZ;
    float* xbuf  = ws + OFF_XBUF;
    float* gate  = ws + OFF_GATE;
    float* dbc   = ws + OFF_DBC;
    float* bcv   = ws + OFF_BCV;
    float* dpre  = ws + OFF_DPRE;
    float* gbuf  = ws + OFF_GBUF;
    float* hbuf  = ws + OFF_H;
    float* krT   = ws + OFF_KRT;
    float* kiT   = ws + OFF_KIT;
    float* pr    = ws + OFF_PR;
    float* pi    = ws + OFF_PI;
    float* probs = ws + OFF_PROBS;
    float* f1    = ws + OFF_F1;

    // 1) modality projections + relu  (WMMA f32)
    launch_gemm(text,   768, W_text,  DMOD, b_text,  reps0, DMOD, NROWS, DMOD, 768, 1, stream);
    launch_gemm(audio,  128, W_audio, DMOD, b_audio, reps1, DMOD, NROWS, DMOD, 128, 1, stream);
    launch_gemm(visual, 512, W_vis,   DMOD, b_vis,   reps2, DMOD, NROWS, DMOD, 512, 1, stream);

    // 2) norm / softmax-weight fusion -> vr  (vi == 0 analytically)
    fuse_vr_kernel<<<NROWS, DMOD, 0, stream>>>(reps0, reps1, reps2, vr);

    // 3) Mamba in-projection: xz = vr @ W_in  [1024 x 512]
    launch_gemm(vr, DMOD, W_in, 2 * DINN, nullptr, xz, 2 * DINN, NROWS, 2 * DINN, DMOD, 0, stream);

    // 4) conv+silu and gate
    xz_post_kernel<<<NROWS, DINN, 0, stream>>>(xz, conv_w, conv_b, xbuf, gate);

    // 5) dbc = x @ W_xproj  [1024 x 40]  (padded ldc = 48, masked cols)
    launch_gemm(xbuf, DINN, W_xproj, DTRANK + 2 * DSTATE, nullptr, dbc, DBC_LD,
                NROWS, DTRANK + 2 * DSTATE, DINN, 0, stream);

    // 6) bc = sum(Bc*Cc)
    bc_kernel<<<NROWS / 256, 256, 0, stream>>>(dbc, bcv);

    // 7) dpre = dt @ W_dt + b_dt   (A = dbc cols 0..7, lda = 48, K = 8)
    launch_gemm(dbc, DBC_LD, W_dt, DINN, b_dt, dpre, DINN, NROWS, DINN, DTRANK, 0, stream);

    // 8) SSM + gating -> gbuf
    ssm_gate_kernel<<<NROWS, DINN, 0, stream>>>(dpre, xbuf, bcv, D_ssm, gate, gbuf);

    // 9) h = gbuf @ W_out  [1024 x 128]
    launch_gemm(gbuf, DINN, W_out, DMOD, nullptr, hbuf, DMOD, NROWS, DMOD, DINN, 0, stream);

    // 10) normalized kernel transposes
    kernel_norm_kernel<<<DMOD, 256, 0, stream>>>(kern, krT, kiT);

    // 11) pr = h @ krT, pi = h @ kiT; probs = pr^2 + pi^2
    launch_gemm(hbuf, DMOD, krT, DMOD, nullptr, pr, DMOD, NROWS, DMOD, DMOD, 0, stream);
    launch_gemm(hbuf, DMOD, kiT, DMOD, nullptr, pi, DMOD, NROWS, DMOD, DMOD, 0, stream);
    probs_sq_kernel<<<(NROWS * DMOD) / 256, 256, 0, stream>>>(pr, pi, probs);

    // 12) f1 = relu(probs @ W_fc1 + b_fc1)  [1024 x 64]
    launch_gemm(probs, DMOD, W_fc1, C1DIM, b_fc1, f1, C1DIM, NROWS, C1DIM, DMOD, 1, stream);

    // 13) head: tanh + log_softmax
    head_kernel<<<NROWS / 256, 256, 0, stream>>>(f1, W_fc2, b_fc2, (float*)d_out);
}